// MaskPolicy_27264452395311
// MI455X (gfx1250) — compile-verified
//
#include <hip/hip_runtime.h>
#include <math.h>

#define BB 2048
#define CC 128
#define LL 16384
#define PP 64
#define ROWS 16          // batches per block (one WMMA M-tile)
#define THREADS 512      // 16 waves of 32
#define SPLIT 4          // L-segments per batch tile

typedef float v2f __attribute__((ext_vector_type(2)));
typedef float v8f __attribute__((ext_vector_type(8)));

__global__ __launch_bounds__(THREADS)
void mask_policy_kernel(const float* __restrict__ logits,
                        const float* __restrict__ u,
                        const int*   __restrict__ shift_x,
                        float* __restrict__ out)   // [B*L] masks, then [B] log_probs
{
    __shared__ float s_logits[CC];
    __shared__ float s_p[CC];
    __shared__ float s_grid[ROWS][CC];
    __shared__ float s_red[CC];
    __shared__ float s_base;

    const int tid  = threadIdx.x;
    const int b0   = blockIdx.x * ROWS;
    const int wid  = tid >> 5;
    const int lane = tid & 31;

    // p = sigmoid(logits); base = sum_c log_sigmoid(-logits[c])
    if (tid < CC) {
        float lg = logits[tid];
        s_logits[tid] = lg;
        s_p[tid]      = 1.0f / (1.0f + expf(-lg));
        s_red[tid]    = -log1pf(expf(lg));        // log_sigmoid(-lg)
    }
    __syncthreads();
    if (tid == 0) {
        float acc = 0.0f;
        for (int c = 0; c < CC; ++c) acc += s_red[c];
        s_base = acc;
    }

    // Bernoulli grid tile (16 x 128) into LDS, coalesced u loads
    #pragma unroll
    for (int i = 0; i < (ROWS * CC) / THREADS; ++i) {
        int e = tid + THREADS * i;
        int r = e >> 7;
        int c = e & (CC - 1);
        float uv = u[(size_t)(b0 + r) * CC + c];
        s_grid[r][c] = (uv < s_p[c]) ? 1.0f : 0.0f;
    }
    __syncthreads();

    // log_probs via V_WMMA_F32_16X16X4_F32: D = grid(16xK) * logits(Kx16 bcast) + D
    if (blockIdx.y == 0 && wid == 0) {
        const int row  = lane & 15;
        const int koff = (lane >> 4) * 2;          // lanes 0-15: K=0,1  lanes 16-31: K=2,3
        v8f acc = {0.f, 0.f, 0.f, 0.f, 0.f, 0.f, 0.f, 0.f};
        #pragma unroll
        for (int kk = 0; kk < CC; kk += 4) {
            v2f a, b;
            a.x = s_grid[row][kk + koff];
            a.y = s_grid[row][kk + koff + 1];
            b.x = s_logits[kk + koff];             // broadcast across all N columns
            b.y = s_logits[kk + koff + 1];
            acc = __builtin_amdgcn_wmma_f32_16x16x4_f32(
                false, a, false, b, (short)0, acc, false, false);
        }
        float base = s_base;
        // D column 0: lane 0 holds M=0..7 in acc[0..7], lane 16 holds M=8..15
        if ((lane & 15) == 0) {
            int mbase = (lane >> 4) * 8;
            #pragma unroll
            for (int i = 0; i < 8; ++i)
                out[(size_t)BB * LL + b0 + mbase + i] = base + acc[i];
        }
    }

    // Mask stream-out: wave w owns batch row w; float4 (b128) coalesced stores
    const int    r     = wid;
    const size_t b     = (size_t)(b0 + r);
    const int    shift = shift_x[b];
    float*       outb  = out + b * LL;
    const float* grow  = s_grid[r];

    const int seg_len = LL / SPLIT;                 // 4096
    const int jbase   = blockIdx.y * seg_len;

    for (int it = 0; it < seg_len / 128; ++it) {
        int j0 = jbase + it * 128 + (lane << 2);
        float4 v;
        #pragma unroll
        for (int q = 0; q < 4; ++q) {
            int j = j0 + q;
            int t = shift + j - PP;                 // index into grid_up after un-pad
            t = (t < 0) ? -t : t;                   // left reflect
            if (t >= LL) t = 2 * LL - 2 - t;        // right reflect
            float pos = fmaxf(((float)t - 63.5f) * 0.0078125f, 0.0f);
            float fi  = floorf(pos);
            int   i0  = (int)fi;                    // in [0,127] by construction
            int   i1  = min(i0 + 1, CC - 1);
            float w   = pos - fi;
            float g0  = grow[i0];
            float g1  = grow[i1];
            float val = g0 + w * (g1 - g0);
            ((float*)&v)[q] = val;
        }
        *(float4*)(outb + j0) = v;
    }
}

extern "C" void kernel_launch(void* const* d_in, const int* in_sizes, int n_in,
                              void* d_out, int out_size, void* d_ws, size_t ws_size,
                              hipStream_t stream) {
    const float* logits = (const float*)d_in[0];
    const float* u      = (const float*)d_in[1];
    const int*   shift  = (const int*)d_in[2];
    float*       out    = (float*)d_out;

    dim3 grid(BB / ROWS, SPLIT);   // 128 batch tiles x 4 L-segments = 512 WGs
    mask_policy_kernel<<<grid, THREADS, 0, stream>>>(logits, u, shift, out);
}